// GNN_10471130267877
// MI455X (gfx1250) — compile-verified
//
#include <hip/hip_runtime.h>
#include <hip/hip_bf16.h>

// ---------------- problem constants (match reference) ----------------
constexpr int NODES   = 100000;   // = 6250 * 16, tile-aligned
constexpr int EDGES   = 1600000;
constexpr int GRAPHS  = 64;       // = 4 * 16
constexpr int DIM     = 128;      // = 8 * 16
constexpr int LAYERS  = 3;
constexpr int CLASSES = 16;       // exactly one WMMA N-tile

typedef float v2f __attribute__((ext_vector_type(2)));
typedef float v8f __attribute__((ext_vector_type(8)));

// ---------------------------------------------------------------------
// Zero-fill (ws is poisoned with 0xAA by the harness)
// ---------------------------------------------------------------------
__global__ void __launch_bounds__(256) zero_kernel(float* __restrict__ p, long long n) {
  for (long long i = blockIdx.x * 256LL + threadIdx.x; i < n;
       i += (long long)gridDim.x * 256LL)
    p[i] = 0.0f;
}

// ---------------------------------------------------------------------
// Edge scatter-add: agg[dst] += x[src].  32 lanes per edge, 4 floats/lane.
// float4 row read is coalesced; atomics land consecutively within a row.
// ---------------------------------------------------------------------
__global__ void __launch_bounds__(256)
scatter_add_kernel(const float* __restrict__ x,
                   const long long* __restrict__ src,
                   const long long* __restrict__ dst,
                   float* __restrict__ agg) {
  const long long total = (long long)EDGES * 32;
  for (long long i = blockIdx.x * 256LL + threadIdx.x; i < total;
       i += (long long)gridDim.x * 256LL) {
    const int e  = (int)(i >> 5);
    const int d4 = (int)(i & 31) << 2;
    const long long s = src[e];
    const long long t = dst[e];
    const float4 v = *(const float4*)(x + s * DIM + d4);
    float* o = agg + t * DIM + d4;
    atomicAdd(o + 0, v.x);
    atomicAdd(o + 1, v.y);
    atomicAdd(o + 2, v.z);
    atomicAdd(o + 3, v.w);
  }
}

// ---------------------------------------------------------------------
// Fused dense layer: xout = relu(agg @ Wrel + brel + xin @ Wroot)
// Block = 4 waves; each wave computes one 16(node) x 16(col) tile with
// V_WMMA_F32_16X16X4_F32, looping K=128 in steps of 4 (2 WMMAs/step).
// Weight slices for this block's N-tile staged in LDS ([k][n] layout,
// conflict-free b32 reads).
// A-matrix fp32 16x4 layout: lanes 0-15 hold K={k0,k0+1}, lanes 16-31
// hold K={k0+2,k0+3} for row M = lane&15.  B 4x16 mirrors this per-VGPR.
// ---------------------------------------------------------------------
__global__ void __launch_bounds__(128)
layer_wmma_kernel(const float* __restrict__ agg,
                  const float* __restrict__ xin,
                  const float* __restrict__ Wrel,
                  const float* __restrict__ brel,
                  const float* __restrict__ Wroot,
                  float* __restrict__ xout) {
  __shared__ float wr[DIM * 16];
  __shared__ float wo[DIM * 16];
  const int nBase = blockIdx.y * 16;

  // cooperative load of both weight slices [128 x 16] for this N-tile
  for (int i = threadIdx.x; i < DIM * 16; i += 128) {
    const int k = i >> 4, n = i & 15;
    wr[i] = Wrel[k * DIM + nBase + n];
    wo[i] = Wroot[k * DIM + nBase + n];
  }
  __syncthreads();

  const int wave  = threadIdx.x >> 5;
  const int lane  = threadIdx.x & 31;
  const int mtile = blockIdx.x * 4 + wave;
  if (mtile >= NODES / 16) return;   // no barriers past this point

  const int m0   = mtile * 16;
  const int half = lane >> 4;        // 0: K lo pair, 1: K hi pair
  const int l15  = lane & 15;
  const int rowA = m0 + l15;
  const int kOff = half * 2;

  const float bias = brel[nBase + l15];
  const float2* aggRow = (const float2*)(agg + (long long)rowA * DIM);
  const float2* xinRow = (const float2*)(xin + (long long)rowA * DIM);

  v8f acc = {0.f, 0.f, 0.f, 0.f, 0.f, 0.f, 0.f, 0.f};

#pragma unroll 4
  for (int k0 = 0; k0 < DIM; k0 += 4) {
    const int kk = k0 + kOff;
    const float2 ar = aggRow[kk >> 1];
    const float2 ax = xinRow[kk >> 1];
    v2f aRel = {ar.x, ar.y};
    v2f aRoot = {ax.x, ax.y};
    v2f bRel  = {wr[kk * 16 + l15], wr[(kk + 1) * 16 + l15]};
    v2f bRoot = {wo[kk * 16 + l15], wo[(kk + 1) * 16 + l15]};
    acc = __builtin_amdgcn_wmma_f32_16x16x4_f32(false, aRel, false, bRel,
                                                (short)0, acc, false, false);
    acc = __builtin_amdgcn_wmma_f32_16x16x4_f32(false, aRoot, false, bRoot,
                                                (short)0, acc, false, false);
  }

  // C/D layout: VGPR i holds M = i (lanes 0-15) or i+8 (lanes 16-31), N = lane&15
#pragma unroll
  for (int i = 0; i < 8; ++i) {
    const int row = m0 + i + half * 8;
    const float v = acc[i] + bias;
    xout[(long long)row * DIM + nBase + l15] = v > 0.f ? v : 0.f;
  }
}

// ---------------------------------------------------------------------
// Per-graph sums + counts (batch ids are sorted int64; atomics into 64x128)
// ---------------------------------------------------------------------
__global__ void __launch_bounds__(256)
pool_kernel(const float* __restrict__ x,
            const long long* __restrict__ batch,
            float* __restrict__ sums,
            float* __restrict__ counts) {
  const long long total = (long long)NODES * 32;
  for (long long i = blockIdx.x * 256LL + threadIdx.x; i < total;
       i += (long long)gridDim.x * 256LL) {
    const int node = (int)(i >> 5);
    const int d4   = (int)(i & 31) << 2;
    const int g    = (int)batch[node];
    const float4 v = *(const float4*)(x + (long long)node * DIM + d4);
    float* o = sums + g * DIM + d4;
    atomicAdd(o + 0, v.x);
    atomicAdd(o + 1, v.y);
    atomicAdd(o + 2, v.z);
    atomicAdd(o + 3, v.w);
    if ((i & 31) == 0) atomicAdd(&counts[g], 1.0f);
  }
}

// ---------------------------------------------------------------------
// Head: out[64,16] = (sums/count) @ lin_w + lin_b.  One block, 4 waves,
// each wave one 16x16 WMMA tile over K=128.
// ---------------------------------------------------------------------
__global__ void __launch_bounds__(128)
head_wmma_kernel(const float* __restrict__ sums,
                 const float* __restrict__ counts,
                 const float* __restrict__ lin_w,
                 const float* __restrict__ lin_b,
                 float* __restrict__ out) {
  const int wave = threadIdx.x >> 5;
  const int lane = threadIdx.x & 31;
  const int half = lane >> 4;
  const int l15  = lane & 15;
  const int m0   = wave * 16;
  const int rowA = m0 + l15;
  const int kOff = half * 2;

  const float inv = 1.0f / fmaxf(counts[rowA], 1.0f);
  const float2* srow = (const float2*)(sums + rowA * DIM);

  v8f acc = {0.f, 0.f, 0.f, 0.f, 0.f, 0.f, 0.f, 0.f};

#pragma unroll 4
  for (int k0 = 0; k0 < DIM; k0 += 4) {
    const int kk = k0 + kOff;
    const float2 s2 = srow[kk >> 1];
    v2f a = {s2.x * inv, s2.y * inv};
    v2f b = {lin_w[kk * CLASSES + l15], lin_w[(kk + 1) * CLASSES + l15]};
    acc = __builtin_amdgcn_wmma_f32_16x16x4_f32(false, a, false, b,
                                                (short)0, acc, false, false);
  }

  const float bias = lin_b[l15];
#pragma unroll
  for (int i = 0; i < 8; ++i)
    out[(m0 + i + half * 8) * CLASSES + l15] = acc[i] + bias;
}

// ---------------------------------------------------------------------
extern "C" void kernel_launch(void* const* d_in, const int* in_sizes, int n_in,
                              void* d_out, int out_size, void* d_ws, size_t ws_size,
                              hipStream_t stream) {
  const float*     x      = (const float*)d_in[0];
  const long long* eidx   = (const long long*)d_in[1];   // [2, EDGES] int64
  const long long* batch  = (const long long*)d_in[2];   // [NODES] int64
  const float*     W_rel  = (const float*)d_in[3];       // [L, D, D]
  const float*     b_rel  = (const float*)d_in[4];       // [L, D]
  const float*     W_root = (const float*)d_in[5];       // [L, D, D]
  const float*     lin_w  = (const float*)d_in[6];       // [D, CLASSES]
  const float*     lin_b  = (const float*)d_in[7];       // [CLASSES]
  float*           outp   = (float*)d_out;

  const long long* src = eidx;
  const long long* dst = eidx + EDGES;

  const long long ND = (long long)NODES * DIM;
  float* bufA   = (float*)d_ws;          // agg scratch (reused)
  float* bufB   = bufA + ND;             // x ping
  float* bufC   = bufB + ND;             // x pong
  float* pooled = bufC + ND;             // [64,128]
  float* counts = pooled + GRAPHS * DIM; // [64]

  const dim3 blk256(256), blk128(128);
  const dim3 gZero(4096), gScat(8192);
  const dim3 gGemm((NODES / 16 + 3) / 4, DIM / 16);  // (1563, 8)

  const float* xin = x;
  float* xout = bufB;

  for (int l = 0; l < LAYERS; ++l) {
    zero_kernel<<<gZero, blk256, 0, stream>>>(bufA, ND);
    scatter_add_kernel<<<gScat, blk256, 0, stream>>>(xin, src, dst, bufA);
    layer_wmma_kernel<<<gGemm, blk128, 0, stream>>>(
        bufA, xin,
        W_rel + (long long)l * DIM * DIM,
        b_rel + (long long)l * DIM,
        W_root + (long long)l * DIM * DIM,
        xout);
    // rotate: next layer reads xout
    if (l == 0)      { xin = bufB; xout = bufC; }
    else if (l == 1) { xin = bufC; xout = bufB; }
  }
  // after L=3: final features are in bufB (l0->B, l1->C, l2->B)

  zero_kernel<<<dim3(32), blk256, 0, stream>>>(pooled, GRAPHS * DIM + GRAPHS);
  pool_kernel<<<gScat, blk256, 0, stream>>>(bufB, batch, pooled, counts);
  head_wmma_kernel<<<dim3(1), blk128, 0, stream>>>(pooled, counts, lin_w, lin_b, outp);
}